// HouseholderLayer_25314537242642
// MI455X (gfx1250) — compile-verified
//
#include <hip/hip_runtime.h>
#include <hip/hip_bf16.h>
#include <stdint.h>

// HouseholderLayer: out = x - 2*(x@vhat)*vhat^T + bias,  vhat = v/(||v||+eps)
// x: 16384 x 4096 fp32. Memory-bound (~0.5 FLOP/B); HBM floor ~536MB/23.3TB/s ~ 23us.
//   pass A (tiny kernel): normalize v -> vhat in d_ws
//   pass B (main): per block of 128 rows:
//     - async-copy vhat into LDS (global_load_async_to_lds_b128)
//     - per wave: 16-row dot products via v_wmma_f32_16x16x4_f32 over LDS-staged
//       16x64 x-chunks, DOUBLE-BUFFERED async DMA (s_wait_asynccnt 8 pipelining)
//     - stream out = x - 2*proj*vhat + bias (b128 NT stores; x re-read hits L2)

typedef __attribute__((ext_vector_type(2))) float v2f;
typedef __attribute__((ext_vector_type(4))) float v4f;
typedef __attribute__((ext_vector_type(8))) float v8f;

#define HH_N           4096
#define HH_N4          (HH_N / 4)
#define HH_BATCH       16384
#define ROWS_PER_BLOCK 128
#define BLOCK_THREADS  256
#define CHUNK          64                  // K-chunk staged in LDS per wave
#define NCHUNKS        (HH_N / CHUNK)      // 64

#if defined(__gfx1250__) && __has_builtin(__builtin_amdgcn_wmma_f32_16x16x4_f32)
#define HH_HAVE_WMMA 1
#else
#define HH_HAVE_WMMA 0
#endif

__device__ __forceinline__ void async_ld_b128(uint32_t lds_off, const float* gsrc) {
  // VDST = LDS byte offset (workgroup-relative), VADDR = 64-bit global address.
  asm volatile("global_load_async_to_lds_b128 %0, %1, off"
               :: "v"(lds_off), "v"((unsigned long long)(uintptr_t)gsrc)
               : "memory");
}

__device__ __forceinline__ void wait_async0() {
#if defined(__gfx1250__) && __has_builtin(__builtin_amdgcn_s_wait_asynccnt)
  __builtin_amdgcn_s_wait_asynccnt(0);
  asm volatile("" ::: "memory");
#else
  asm volatile("s_wait_asynccnt 0" ::: "memory");
#endif
}

__device__ __forceinline__ void wait_async_le8() {
  // Async loads complete in order: <=8 outstanding guarantees the previous
  // chunk's 8 b128 transfers have landed while the next chunk is in flight.
#if defined(__gfx1250__) && __has_builtin(__builtin_amdgcn_s_wait_asynccnt)
  __builtin_amdgcn_s_wait_asynccnt(8);
  asm volatile("" ::: "memory");
#else
  asm volatile("s_wait_asynccnt 8" ::: "memory");
#endif
}

// ---------------- pass A: vhat = v / (||v|| + eps) ----------------
__global__ __launch_bounds__(BLOCK_THREADS)
void hh_normalize(const float* __restrict__ v, float* __restrict__ vhat) {
  __shared__ float red[BLOCK_THREADS];
  const int tid = threadIdx.x;
  float s = 0.0f;
  for (int i = tid; i < HH_N; i += BLOCK_THREADS) {
    float t = v[i];
    s += t * t;
  }
  red[tid] = s;
  __syncthreads();
  for (int off = BLOCK_THREADS / 2; off > 0; off >>= 1) {
    if (tid < off) red[tid] += red[tid + off];
    __syncthreads();
  }
  const float inv = 1.0f / (sqrtf(red[0]) + 1e-14f);
  for (int i = tid; i < HH_N; i += BLOCK_THREADS) vhat[i] = v[i] * inv;
}

// ---------------- pass B: main streaming kernel ----------------
__global__ __launch_bounds__(BLOCK_THREADS)
void hh_main(const float* __restrict__ x, const float* __restrict__ vhat_g,
             const float* __restrict__ bias, float* __restrict__ out) {
  __shared__ float s_vhat[HH_N];                   // 16 KB
  __shared__ float s_x[8][2][16 * CHUNK];          // 8 waves * 2 bufs * 4 KB = 64 KB
  __shared__ float s_proj[ROWS_PER_BLOCK];         // 512 B

  const int tid  = threadIdx.x;
  const int wave = tid >> 5;
  const int lane = tid & 31;
  const int row0 = blockIdx.x * ROWS_PER_BLOCK;

  // --- stage vhat into LDS via async DMA ---
  for (int i = tid * 4; i < HH_N; i += BLOCK_THREADS * 4) {
    async_ld_b128((uint32_t)(uintptr_t)&s_vhat[i], vhat_g + i);
  }
  wait_async0();
  __syncthreads();

  // --- per-wave 16-row dot products: proj[m] = sum_k x[m,k] * vhat[k] ---
  const int wr0  = row0 + wave * 16;
  const int m    = lane & 15;   // row within wave tile
  const int half = lane >> 4;   // 0 / 1
  const int kq   = half * 2;    // K sub-offset: lanes 0-15 -> K 0,1; 16-31 -> K 2,3

  // Lane covers row m, cols [half*32, half*32+32) of each chunk: 8 x b128.
  const float* rowbase = x + (size_t)(wr0 + m) * HH_N + half * 32;
  const uint32_t dst0 =
      (uint32_t)(uintptr_t)&s_x[wave][0][m * CHUNK + half * 32];
  const uint32_t dst1 =
      (uint32_t)(uintptr_t)&s_x[wave][1][m * CHUNK + half * 32];

  auto issue_chunk = [&](int k0, int buf) {
    const float* src = rowbase + k0;
    const uint32_t dst = buf ? dst1 : dst0;
#pragma unroll
    for (int i = 0; i < 8; ++i) {
      async_ld_b128(dst + (uint32_t)(i * 16), src + i * 4);
    }
  };

#if HH_HAVE_WMMA
  v8f c = {};
#else
  float acc = 0.0f;
#endif

  issue_chunk(0, 0);                         // prefetch chunk 0
  for (int ci = 0; ci < NCHUNKS; ++ci) {
    const int buf = ci & 1;
    if (ci + 1 < NCHUNKS) {
      issue_chunk((ci + 1) * CHUNK, buf ^ 1);  // prefetch next chunk
      wait_async_le8();                        // oldest 8 (current chunk) done
    } else {
      wait_async0();
    }
    const float* sx = &s_x[wave][buf][0];
    const int k0 = ci * CHUNK;

#pragma unroll
    for (int kk = 0; kk < CHUNK; kk += 4) {
      v2f a, b;
      a.x = sx[m * CHUNK + kk + kq];
      a.y = sx[m * CHUNK + kk + kq + 1];
      b.x = s_vhat[k0 + kk + kq];        // B[k,n] = vhat[k], replicated over N
      b.y = s_vhat[k0 + kk + kq + 1];
#if HH_HAVE_WMMA
      c = __builtin_amdgcn_wmma_f32_16x16x4_f32(
          /*neg_a=*/false, a, /*neg_b=*/false, b,
          /*c_mod=*/(short)0, c, /*reuse_a=*/false, /*reuse_b=*/false);
#else
      acc += a.x * b.x + a.y * b.y;
#endif
    }
  }

#if HH_HAVE_WMMA
  // D[m,n] == proj[m] for every n. Lane 0 holds M=0..7 in c[0..7], lane 16 M=8..15.
  if (m == 0) {
    float* dp = &s_proj[wave * 16 + half * 8];
#pragma unroll
    for (int i = 0; i < 8; ++i) dp[i] = c[i];
  }
#else
  acc += __shfl_xor(acc, 16, 32);   // combine K%4 in {0,1} with {2,3}
  if (half == 0) s_proj[wave * 16 + m] = acc;
#endif
  __syncthreads();

  // --- stream the rank-1 update: out = x - 2*proj*vhat + bias ---
  // x second read hits L2 (just streamed through it); out is write-once ->
  // non-temporal b128 stores so 268MB of output doesn't evict x/bias from L2.
  const v4f* x4 = (const v4f*)x + (size_t)row0 * HH_N4;
  const v4f* b4 = (const v4f*)bias;
  const v4f* v4 = (const v4f*)s_vhat;
  v4f* o4 = (v4f*)out + (size_t)row0 * HH_N4;

  for (int idx = tid; idx < ROWS_PER_BLOCK * HH_N4; idx += BLOCK_THREADS) {
    const int r  = idx >> 10;           // HH_N4 == 1024
    const int c4 = idx & (HH_N4 - 1);
    const v4f xv = x4[idx];
    const float p2 = 2.0f * s_proj[r];
    const v4f vh = v4[c4];
    const v4f bb = b4[c4];
    const v4f ov = xv - p2 * vh + bb;
    __builtin_nontemporal_store(ov, &o4[idx]);
  }
}

extern "C" void kernel_launch(void* const* d_in, const int* in_sizes, int n_in,
                              void* d_out, int out_size, void* d_ws, size_t ws_size,
                              hipStream_t stream) {
  (void)in_sizes; (void)n_in; (void)out_size; (void)ws_size;
  const float* x    = (const float*)d_in[0];
  const float* vec  = (const float*)d_in[1];
  const float* bias = (const float*)d_in[2];
  float* out  = (float*)d_out;
  float* vhat = (float*)d_ws;   // 4096 floats of scratch

  hh_normalize<<<1, BLOCK_THREADS, 0, stream>>>(vec, vhat);
  hh_main<<<HH_BATCH / ROWS_PER_BLOCK, BLOCK_THREADS, 0, stream>>>(x, vhat, bias, out);
}